// VanillaModularityLoss_44220983280093
// MI455X (gfx1250) — compile-verified
//
#include <hip/hip_runtime.h>
#include <hip/hip_bf16.h>

typedef float v2f __attribute__((ext_vector_type(2)));
typedef float v8f __attribute__((ext_vector_type(8)));

static constexpr int NN    = 16384;
static constexpr int KD    = 16;
static constexpr int TILES = NN / 16;     // 1024 row tiles
static constexpr int WPB   = 8;           // waves per block (256 threads, wave32)
static constexpr int CHUNK = NN / WPB;    // 2048 reduction columns per wave
static constexpr int STEP  = 16;          // columns per pipeline stage (4 WMMAs)
static constexpr int ITERS = CHUNK / STEP;  // 128 stages (multiple of 4)

// Fragment set for one pipeline stage: 4 A-fragments (adj) + 4 B-fragments (Ct)
struct Frag {
    v2f a0, a1, a2, a3;
    v2f b0, b1, b2, b3;
};

__device__ __forceinline__ void load_frag(Frag& f, const float* ap, const float* bp)
{
    f.a0 = __builtin_nontemporal_load((const v2f*)(ap));
    f.a1 = __builtin_nontemporal_load((const v2f*)(ap + 4));
    f.a2 = __builtin_nontemporal_load((const v2f*)(ap + 8));
    f.a3 = __builtin_nontemporal_load((const v2f*)(ap + 12));
    f.b0 = *(const v2f*)(bp);
    f.b1 = *(const v2f*)(bp + 4);
    f.b2 = *(const v2f*)(bp + 8);
    f.b3 = *(const v2f*)(bp + 12);
}

__device__ __forceinline__ void compute_frag(const Frag& f,
                                             v8f& c0, v8f& c1, v8f& c2, v8f& c3,
                                             v2f& d0, v2f& d1, v2f& d2, v2f& d3)
{
    c0 = __builtin_amdgcn_wmma_f32_16x16x4_f32(false, f.a0, false, f.b0, (short)0, c0, false, false);
    c1 = __builtin_amdgcn_wmma_f32_16x16x4_f32(false, f.a1, false, f.b1, (short)0, c1, false, false);
    c2 = __builtin_amdgcn_wmma_f32_16x16x4_f32(false, f.a2, false, f.b2, (short)0, c2, false, false);
    c3 = __builtin_amdgcn_wmma_f32_16x16x4_f32(false, f.a3, false, f.b3, (short)0, c3, false, false);
    d0 += f.a0;  d1 += f.a1;  d2 += f.a2;  d3 += f.a3;   // deg = adj @ 1 partials
}

// ---------------------------------------------------------------------------
// Kernel 0: transpose C (N x 16, row-major) -> Ct (16 x N). Makes the WMMA
// B-fragment a single contiguous b64 load per lane (same shape as A loads).
// 1 MB total; stays resident in the 192 MB L2 across all 1024 tile blocks.
// ---------------------------------------------------------------------------
__global__ __launch_bounds__(256)
void transposeC(const float* __restrict__ C, float* __restrict__ Ct)
{
    const int c = blockIdx.x * 256 + threadIdx.x;   // 16384 threads, one column
#pragma unroll
    for (int n = 0; n < KD; ++n)
        Ct[(size_t)n * NN + c] = C[(size_t)c * KD + n];  // writes coalesced
}

// ---------------------------------------------------------------------------
// Kernel 1: one block per 16-row tile of adj. Each wave accumulates a
// 16x16 f32 tile of AC = adj @ C over its 2048-column reduction slice using
// V_WMMA_F32_16X16X4_F32 (4 accumulator chains). Software-pipelined with a
// 4-buffer rotation at prefetch distance 2: ~16 b64 loads always in flight.
// ---------------------------------------------------------------------------
__global__ __launch_bounds__(256, 2)
void modloss_tile(const float* __restrict__ C,
                  const float* __restrict__ Ct,
                  const float* __restrict__ adj,
                  float* __restrict__ ws_sumCAC,   // [TILES]
                  float* __restrict__ ws_total,    // [TILES]
                  float* __restrict__ ws_dC,       // [TILES*16]
                  float* __restrict__ ws_colsum)   // [TILES*16]
{
    __shared__ float lds_acc[WPB * 256];
    __shared__ float lds_deg[WPB * 32];
    __shared__ float lds_red[256];
    __shared__ float s_deg[16];

    const int tile  = blockIdx.x;
    const int rbase = tile << 4;
    const int tid   = threadIdx.x;
    const int wave  = tid >> 5;
    const int lane  = tid & 31;
    const int half  = lane >> 4;   // K-half select for A/B fragments
    const int row   = lane & 15;   // A-matrix row (M) / B-matrix column (N)

    // Per-lane streaming pointers; fragment layout: v0 -> K=2h, v1 -> K=2h+1
    const float* aptr = adj + (size_t)(rbase + row) * NN + wave * CHUNK + 2 * half;
    const float* bptr = Ct  + (size_t)row * NN          + wave * CHUNK + 2 * half;

    v8f acc0 = {}, acc1 = {}, acc2 = {}, acc3 = {};
    v2f dv0 = {}, dv1 = {}, dv2 = {}, dv3 = {};

    Frag f0, f1, f2, f3;

    // Prologue: two stages in flight before any compute.
    load_frag(f0, aptr,        bptr);
    load_frag(f1, aptr + STEP, bptr + STEP);

    // Steady state: 4 stages per trip; every buffer is loaded two compute
    // slots before its WMMAs consume it (prefetch distance 2, no copies).
    // Loads reach stage i+5, so run while i+5 <= ITERS-1.
    int i = 0;
#pragma unroll 1
    for (; i + 6 <= ITERS; i += 4) {
        load_frag(f2, aptr + 2 * STEP, bptr + 2 * STEP);
        compute_frag(f0, acc0, acc1, acc2, acc3, dv0, dv1, dv2, dv3);
        load_frag(f3, aptr + 3 * STEP, bptr + 3 * STEP);
        compute_frag(f1, acc0, acc1, acc2, acc3, dv0, dv1, dv2, dv3);
        load_frag(f0, aptr + 4 * STEP, bptr + 4 * STEP);
        compute_frag(f2, acc0, acc1, acc2, acc3, dv0, dv1, dv2, dv3);
        load_frag(f1, aptr + 5 * STEP, bptr + 5 * STEP);
        compute_frag(f3, acc0, acc1, acc2, acc3, dv0, dv1, dv2, dv3);
        aptr += 4 * STEP;
        bptr += 4 * STEP;
    }
    // Epilogue: i == ITERS-4. f0,f1 hold stages ITERS-4, ITERS-3; the last
    // two stages are loaded here — never past the wave's slice.
    load_frag(f2, aptr + 2 * STEP, bptr + 2 * STEP);
    compute_frag(f0, acc0, acc1, acc2, acc3, dv0, dv1, dv2, dv3);
    load_frag(f3, aptr + 3 * STEP, bptr + 3 * STEP);
    compute_frag(f1, acc0, acc1, acc2, acc3, dv0, dv1, dv2, dv3);
    compute_frag(f2, acc0, acc1, acc2, acc3, dv0, dv1, dv2, dv3);
    compute_frag(f3, acc0, acc1, acc2, acc3, dv0, dv1, dv2, dv3);

    const v2f dv = (dv0 + dv1) + (dv2 + dv3);
#pragma unroll
    for (int v = 0; v < 8; ++v)
        lds_acc[wave * 256 + v * 32 + lane] = (acc0[v] + acc1[v]) + (acc2[v] + acc3[v]);
    lds_deg[wave * 32 + lane] = dv.x + dv.y;
    __syncthreads();

    // Combine the 8 wave-partial AC tiles; contract with C immediately.
    {
        const int v  = tid >> 5;
        const int ln = tid & 31;
        const int M  = v + 8 * (ln >> 4);     // C/D fragment layout
        const int n  = ln & 15;
        float ac = 0.f;
#pragma unroll
        for (int w = 0; w < WPB; ++w) ac += lds_acc[w * 256 + tid];
        lds_red[tid] = ac * C[(size_t)(rbase + M) * KD + n];
    }
    if (tid < 16) {                            // full row sums for this tile
        float d = 0.f;
#pragma unroll
        for (int w = 0; w < WPB; ++w)
            d += lds_deg[w * 32 + tid] + lds_deg[w * 32 + 16 + tid];
        s_deg[tid] = d;
    }
    __syncthreads();

    for (int s = 128; s > 0; s >>= 1) {
        if (tid < s) lds_red[tid] += lds_red[tid + s];
        __syncthreads();
    }
    if (tid == 0) ws_sumCAC[tile] = lds_red[0];

    if (tid < 16) {
        const int k = tid;
        float dck = 0.f, cs = 0.f;
#pragma unroll
        for (int r = 0; r < 16; ++r) {
            const float cv = C[(size_t)(rbase + r) * KD + k];
            dck += s_deg[r] * cv;              // (deg @ C) partial
            cs  += cv;                         // column-sum partial
        }
        ws_dC[tile * 16 + k]     = dck;
        ws_colsum[tile * 16 + k] = cs;
        if (k == 0) {
            float t = 0.f;
#pragma unroll
            for (int r = 0; r < 16; ++r) t += s_deg[r];
            ws_total[tile] = t;                // partial of 2m = sum(deg)
        }
    }
}

// ---------------------------------------------------------------------------
// Kernel 2: deterministic reduction of the 1024 tile partials + final scalar.
// ---------------------------------------------------------------------------
__global__ __launch_bounds__(256)
void modloss_final(const float* __restrict__ ws_sumCAC,
                   const float* __restrict__ ws_total,
                   const float* __restrict__ ws_dC,
                   const float* __restrict__ ws_colsum,
                   const float* __restrict__ beta_p,
                   float* __restrict__ out)
{
    __shared__ float red[256];
    __shared__ float s_sum[2];
    __shared__ float s_dC[16];
    __shared__ float s_cs[16];
    const int tid = threadIdx.x;

    float p1 = 0.f, p2 = 0.f;
    for (int t = tid; t < TILES; t += 256) { p1 += ws_sumCAC[t]; p2 += ws_total[t]; }
    red[tid] = p1; __syncthreads();
    for (int s = 128; s > 0; s >>= 1) { if (tid < s) red[tid] += red[tid + s]; __syncthreads(); }
    if (tid == 0) s_sum[0] = red[0];
    __syncthreads();
    red[tid] = p2; __syncthreads();
    for (int s = 128; s > 0; s >>= 1) { if (tid < s) red[tid] += red[tid + s]; __syncthreads(); }
    if (tid == 0) s_sum[1] = red[0];
    __syncthreads();

    const int k = tid & 15;
    const int g = tid >> 4;                    // 16 groups x 16 k-slots
    float pd = 0.f, pc = 0.f;
    for (int t = g; t < TILES; t += 16) { pd += ws_dC[t * 16 + k]; pc += ws_colsum[t * 16 + k]; }
    red[tid] = pd; __syncthreads();
    for (int s = 8; s > 0; s >>= 1) { if (g < s) red[g * 16 + k] += red[(g + s) * 16 + k]; __syncthreads(); }
    if (g == 0) s_dC[k] = red[k];
    __syncthreads();
    red[tid] = pc; __syncthreads();
    for (int s = 8; s > 0; s >>= 1) { if (g < s) red[g * 16 + k] += red[(g + s) * 16 + k]; __syncthreads(); }
    if (g == 0) s_cs[k] = red[k];
    __syncthreads();

    if (tid == 0) {
        const float sumCAC = s_sum[0];
        const float twoM   = s_sum[1];         // 2m = sum(deg)
        float dot = 0.f, col = 0.f;
#pragma unroll
        for (int i = 0; i < 16; ++i) {
            dot += s_dC[i] * s_dC[i];
            col += fabsf(s_cs[i] - 1.0f);
        }
        const float modularity = sumCAC - dot / twoM;
        const float mod_loss   = -modularity / twoM;
        const float collapse   = (4.0f / (float)NN) * col;   // sqrt(16)/N
        out[0] = mod_loss + beta_p[0] * collapse;
    }
}

extern "C" void kernel_launch(void* const* d_in, const int* in_sizes, int n_in,
                              void* d_out, int out_size, void* d_ws, size_t ws_size,
                              hipStream_t stream)
{
    const float* C    = (const float*)d_in[0];
    // d_in[1] = X, unused by the loss
    const float* adj  = (const float*)d_in[2];
    const float* beta = (const float*)d_in[3];

    float* ws        = (float*)d_ws;
    float* Ct        = ws;                             // 16*NN floats (1 MB)
    float* ws_sumCAC = ws + (size_t)KD * NN;           // TILES floats
    float* ws_total  = ws_sumCAC + TILES;              // TILES floats
    float* ws_dC     = ws_total + TILES;               // TILES*16 floats
    float* ws_colsum = ws_dC + TILES * KD;             // TILES*16 floats

    transposeC<<<NN / 256, 256, 0, stream>>>(C, Ct);
    modloss_tile<<<TILES, 256, 0, stream>>>(C, Ct, adj,
                                            ws_sumCAC, ws_total, ws_dC, ws_colsum);
    modloss_final<<<1, 256, 0, stream>>>(ws_sumCAC, ws_total, ws_dC, ws_colsum,
                                         beta, (float*)d_out);
}